// YOLOLossv3_37134287241289
// MI455X (gfx1250) — compile-verified
//
#include <hip/hip_runtime.h>
#include <hip/hip_bf16.h>

// YOLOv3 loss for MI455X (gfx1250, wave32).
// Structure: (1) init masks+accumulators, (2) scatter 240 GTs into obj/noobj
// masks + target records, (3) sparse main pass (conf channel everywhere,
// full 85 channels only at obj cells) with per-wave f32 WMMA reduction,
// (4) scalar finalize. All f32; traffic ~2.6 MB -> launch-latency bound.

#define NGRID 52
#define NANCH 3
#define NCLS  80
#define PLANE (NGRID * NGRID)        // 2704
#define CELLS_PER_B (NANCH * PLANE)  // 8112
#define CHN (NCLS + 5)               // 85

typedef __attribute__((ext_vector_type(2))) float v2f;
typedef __attribute__((ext_vector_type(8))) float v8f;

__device__ __forceinline__ float sigm(float z)   { return 1.0f / (1.0f + expf(-z)); }
__device__ __forceinline__ float clogp(float p)  { return fmaxf(logf(p),    -100.0f); }
__device__ __forceinline__ float clog1m(float p) { return fmaxf(log1pf(-p), -100.0f); }

__global__ void yolo_init(int* __restrict__ objIdx, int* __restrict__ noobj,
                          float* __restrict__ accum, int nCells) {
    int t = blockIdx.x * blockDim.x + threadIdx.x;
    if (t < nCells) { objIdx[t] = 0; noobj[t] = 1; }
    if (t < 16) accum[t] = 0.0f;
}

__global__ void yolo_scatter(const float* __restrict__ gts, int nGts,
                             int* __restrict__ objIdx, int* __restrict__ noobj,
                             float* __restrict__ gtT, int* __restrict__ gtLab) {
    int g = blockIdx.x * blockDim.x + threadIdx.x;
    if (g >= nGts) return;
    const float awv[3] = {0.05f, 0.12f, 0.28f};
    const float ahv[3] = {0.07f, 0.16f, 0.35f};
    float bF  = gts[g * 6 + 0];
    float lbF = gts[g * 6 + 1];
    float gx  = gts[g * 6 + 2], gy = gts[g * 6 + 3];
    float gw  = gts[g * 6 + 4], gh = gts[g * 6 + 5];
    int bi = (int)bF;
    float iou[3];
    #pragma unroll
    for (int a = 0; a < 3; a++) {
        float inter = fminf(gw, awv[a]) * fminf(gh, ahv[a]);
        iou[a] = inter / (gw * gh + awv[a] * ahv[a] - inter);
    }
    int best = 0;                       // first-max semantics like jnp.argmax
    if (iou[1] > iou[best]) best = 1;
    if (iou[2] > iou[best]) best = 2;
    float gxs = gx * (float)NGRID, gys = gy * (float)NGRID;
    int gi = (int)gxs, gj = (int)gys;
    int cellBase = bi * CELLS_PER_B + gj * NGRID + gi;
    #pragma unroll
    for (int a = 0; a < 3; a++) {       // keep = (iou <= 0.5); clear when !keep or best
        if (iou[a] > 0.5f || a == best) noobj[cellBase + a * PLANE] = 0;
    }
    atomicMax(&objIdx[cellBase + best * PLANE], g + 1);  // max-g wins ties (consistent tb/tcls)
    gtT[g * 4 + 0] = gxs - floorf(gxs);
    gtT[g * 4 + 1] = gys - floorf(gys);
    gtT[g * 4 + 2] = logf(gw / awv[best]);
    gtT[g * 4 + 3] = logf(gh / ahv[best]);
    gtLab[g] = (int)lbF;
}

__global__ __launch_bounds__(256) void yolo_main(
    const float* __restrict__ out, const int* __restrict__ objIdx,
    const int* __restrict__ noobj, const float* __restrict__ gtT,
    const int* __restrict__ gtLab, float* __restrict__ accum, int nCells) {
    int t = blockIdx.x * blockDim.x + threadIdx.x;

    // vals: {Sx, Sy, Sw, Sh, Sconf_obj, Sconf_noobj, Scls, Nobj, Nnoobj}
    float vals[9] = {0.f, 0.f, 0.f, 0.f, 0.f, 0.f, 0.f, 0.f, 0.f};

    if (t < nCells) {
        int i = t % NGRID;
        int j = (t / NGRID) % NGRID;
        int a = (t / PLANE) % NANCH;
        int b = t / CELLS_PER_B;
        int base = ((b * NANCH + a) * CHN) * PLANE + j * NGRID + i;
        int oi = objIdx[t];
        int nb = noobj[t];
        if (oi > 0 || nb != 0) {
            float pc = sigm(out[base + 4 * PLANE]);      // conf channel (coalesced along i)
            if (oi > 0)  { vals[4] = -clogp(pc);  vals[7] = 1.0f; }  // t=1 branch of BCE
            if (nb != 0) { vals[5] = -clog1m(pc); vals[8] = 1.0f; }  // t=0 branch of BCE
        }
        if (oi > 0) {                                    // <=240 threads take this path
            int g = oi - 1;
            float tx = gtT[g * 4 + 0], ty = gtT[g * 4 + 1];
            float lw = gtT[g * 4 + 2], lh = gtT[g * 4 + 3];
            int lab = gtLab[g];
            float dx = sigm(out[base + 0 * PLANE]) - tx;
            float dy = sigm(out[base + 1 * PLANE]) - ty;
            float dw = out[base + 2 * PLANE] - lw;
            float dh = out[base + 3 * PLANE] - lh;
            vals[0] = dx * dx; vals[1] = dy * dy;
            vals[2] = dw * dw; vals[3] = dh * dh;
            float cl = 0.0f;
            for (int c = 0; c < NCLS; c++) {
                float p = sigm(out[base + (5 + c) * PLANE]);
                cl += (c == lab) ? -clogp(p) : -clog1m(p);
            }
            vals[6] = cl;
        }
    }

    // ---- per-wave 32-lane reduction via V_WMMA_F32_16X16X4_F32 ----
    // A = {partial, 0} per lane (16x4 f32), B = ones (4x16) => D[m][n] = rowsum_m(A).
    // Sum of 8 D regs per lane covers half the rows; xor-16 shuffle adds the other half.
    __shared__ float waveTot[8][9];
    int wave = threadIdx.x >> 5;
    int lane = threadIdx.x & 31;
    #pragma unroll
    for (int r = 0; r < 9; r++) {
        v2f A = {vals[r], 0.0f};
        v2f B = {1.0f, 1.0f};
        v8f C = {};
        v8f D = __builtin_amdgcn_wmma_f32_16x16x4_f32(
            /*neg_a=*/false, A, /*neg_b=*/false, B,
            /*c_mod=*/(short)0, C, /*reuse_a=*/false, /*reuse_b=*/false);
        float s = D[0] + D[1] + D[2] + D[3] + D[4] + D[5] + D[6] + D[7];
        s += __shfl_xor(s, 16, 32);
        if (lane == 0) waveTot[wave][r] = s;
    }
    __syncthreads();
    if (threadIdx.x < 9) {
        float tot = 0.0f;
        #pragma unroll
        for (int w = 0; w < 8; w++) tot += waveTot[w][threadIdx.x];
        atomicAdd(&accum[threadIdx.x], tot);
    }
}

__global__ void yolo_final(const float* __restrict__ accum, float* __restrict__ outp) {
    if (threadIdx.x == 0 && blockIdx.x == 0) {
        float Sx = accum[0], Sy = accum[1], Sw = accum[2], Sh = accum[3];
        float Sco = accum[4], Scn = accum[5], Scls = accum[6];
        float No = accum[7], Nn = accum[8];
        float dO = fmaxf(No, 1.0f);
        float dN = fmaxf(Nn, 1.0f);
        float dC = fmaxf((float)NCLS * No, 1.0f);
        outp[0] = (Sx + Sy) / dO + (Sw + Sh) / dO
                + 1.0f * (Sco / dO) + 100.0f * (Scn / dN) + Scls / dC;
    }
}

extern "C" void kernel_launch(void* const* d_in, const int* in_sizes, int n_in,
                              void* d_out, int out_size, void* d_ws, size_t ws_size,
                              hipStream_t stream) {
    (void)n_in; (void)out_size; (void)ws_size;
    const float* out = (const float*)d_in[0];
    const float* gts = (const float*)d_in[1];
    int nGts = in_sizes[1] / 6;
    int nB = in_sizes[0] / (NANCH * CHN * PLANE);
    int nCells = nB * CELLS_PER_B;

    char* ws = (char*)d_ws;
    float* accum = (float*)ws;                                      // 16 f32
    int*   objIdx = (int*)(ws + 64);                                // nCells i32
    int*   noobj  = (int*)(ws + 64 + (size_t)nCells * 4);           // nCells i32
    float* gtT    = (float*)(ws + 64 + (size_t)nCells * 8);         // nGts x 4 f32
    int*   gtLab  = (int*)(ws + 64 + (size_t)nCells * 8 + (size_t)nGts * 16);

    int blocks = (nCells + 255) / 256;
    yolo_init<<<blocks, 256, 0, stream>>>(objIdx, noobj, accum, nCells);
    yolo_scatter<<<(nGts + 255) / 256, 256, 0, stream>>>(gts, nGts, objIdx, noobj, gtT, gtLab);
    yolo_main<<<blocks, 256, 0, stream>>>(out, objIdx, noobj, gtT, gtLab, accum, nCells);
    yolo_final<<<1, 32, 0, stream>>>(accum, (float*)d_out);
}